// SSAVF_9801115369932
// MI455X (gfx1250) — compile-verified
//
#include <hip/hip_runtime.h>
#include <hip/hip_bf16.h>

// ---------------- problem constants ----------------
#define T_DIM   52
#define B_DIM   4
#define C_DIM   9
#define HW      4096            // 64*64
#define N_PIX   16384           // b*h*w
#define F_DIM   64
#define NCELL   6
#define TH_D    5
#define NSTEPS  100
#define LN_EPS  1e-5f
#define ZLD     66              // padded row stride for per-wave z tiles

typedef float v2f __attribute__((ext_vector_type(2)));
typedef float v8f __attribute__((ext_vector_type(8)));
typedef unsigned int v4u __attribute__((ext_vector_type(4)));
typedef int v4i __attribute__((ext_vector_type(4)));
typedef int v8i __attribute__((ext_vector_type(8)));

__device__ __forceinline__ v8f wmma4(v2f a, v2f b, v8f c) {
    // D = A(16x4 f32) * B(4x16 f32) + C(16x16 f32), wave32
    return __builtin_amdgcn_wmma_f32_16x16x4_f32(false, a, false, b, (short)0, c,
                                                 false, false);
}

__device__ __forceinline__ int iclamp(int v, int lo, int hi) {
    return v < lo ? lo : (v > hi ? hi : v);
}

// ---------------- kernel 1: CPAB null-space basis (orthonormal, Gram-Schmidt) ----
__global__ void k_basis(float* bbuf) {
    if (threadIdx.x != 0 || blockIdx.x != 0) return;
    float M[12][5];
    // hat-function parametrization of continuous piecewise-linear v with v(0)=v(1)=0
    for (int k = 0; k < 5; ++k) {
        float u[7];
        #pragma unroll
        for (int j = 0; j < 7; ++j) u[j] = 0.f;
        u[k + 1] = 1.f;
        for (int i = 0; i < 6; ++i) {
            float a = 6.f * (u[i + 1] - u[i]);
            float b = u[i] - (float)i * (u[i + 1] - u[i]);
            M[2 * i][k]     = a;
            M[2 * i + 1][k] = b;
        }
    }
    // modified Gram-Schmidt -> orthonormal columns
    for (int k = 0; k < 5; ++k) {
        for (int j = 0; j < k; ++j) {
            float d = 0.f;
            for (int i = 0; i < 12; ++i) d += M[i][j] * M[i][k];
            for (int i = 0; i < 12; ++i) M[i][k] -= d * M[i][j];
        }
        float nrm = 0.f;
        for (int i = 0; i < 12; ++i) nrm += M[i][k] * M[i][k];
        nrm = rsqrtf(nrm);
        for (int i = 0; i < 12; ++i) M[i][k] *= nrm;
    }
    for (int i = 0; i < 12; ++i)
        for (int k = 0; k < 5; ++k) bbuf[i * 5 + k] = M[i][k];
}

// ---------------- kernel 2: theta -> per-cell affine params A[N,6,2] ------------
__global__ void k_theta2A(const float* __restrict__ th_in, float scale,
                          const float* __restrict__ bbuf,
                          float* __restrict__ Abuf, float* __restrict__ th_out) {
    int n = blockIdx.x * blockDim.x + threadIdx.x;
    if (n >= N_PIX) return;
    float th[TH_D];
    #pragma unroll
    for (int j = 0; j < TH_D; ++j) th[j] = scale * th_in[n * TH_D + j];
    if (th_out) {
        #pragma unroll
        for (int j = 0; j < TH_D; ++j) th_out[n * TH_D + j] = th[j];
    }
    #pragma unroll
    for (int m = 0; m < 12; ++m) {
        float s = 0.f;
        #pragma unroll
        for (int j = 0; j < TH_D; ++j) s += bbuf[m * 5 + j] * th[j];
        Abuf[n * 12 + m] = s;
    }
}

// ---------------- kernel 3: CPAB ODE + 1D interpolation -------------------------
// x: [t,b,c,h,w]; Abuf: [N,12]; out: [N,c,t]
__global__ void k_cpab(const float* __restrict__ x, const float* __restrict__ Abuf,
                       float* __restrict__ out) {
    int id = blockIdx.x * blockDim.x + threadIdx.x;
    if (id >= N_PIX * T_DIM) return;
    int n  = id / T_DIM;
    int tt = id - n * T_DIM;
    float A[12];
    #pragma unroll
    for (int m = 0; m < 12; ++m) A[m] = Abuf[n * 12 + m];

    float xc = (float)tt * (1.f / (T_DIM - 1));
    const float h = 1.f / NSTEPS;
    #pragma unroll 4
    for (int s = 0; s < NSTEPS; ++s) {
        int idx = iclamp((int)(xc * (float)NCELL), 0, NCELL - 1);
        xc = fminf(fmaxf(xc + h * (A[2 * idx] * xc + A[2 * idx + 1]), 0.f), 1.f);
    }
    float pos = xc * (float)(T_DIM - 1);
    int   x0  = iclamp((int)floorf(pos), 0, T_DIM - 2);
    float w   = pos - (float)x0;

    int bb = n >> 12;          // batch
    int pg = n & 4095;         // pixel in 64x64 plane
    #pragma unroll
    for (int ch = 0; ch < C_DIM; ++ch) {
        float d0 = x[(((x0)     * B_DIM + bb) * C_DIM + ch) * HW + pg];
        float d1 = x[(((x0 + 1) * B_DIM + bb) * C_DIM + ch) * HW + pg];
        out[(n * C_DIM + ch) * T_DIM + tt] = d0 * (1.f - w) + d1 * w;
    }
}

// ---------------- kernel 4: fused localise (x2) + fc1 accumulation --------------
// grid: (4 batches, 32 pixel-tiles of 128), block 256 = 8 waves, wave owns 16 pixels.
// acc out: [N,16]
__global__ void __launch_bounds__(256)
k_loc_fc(const float* __restrict__ x, const float* __restrict__ xs,
         const float* __restrict__ lnw_in, const float* __restrict__ lnb_in,
         const float* __restrict__ w1, const float* __restrict__ b1,
         const float* __restrict__ w2, const float* __restrict__ b2,
         const float* __restrict__ lnw_c, const float* __restrict__ lnb_c,
         const float* __restrict__ wfc, float* __restrict__ acc) {
    __shared__ float s_w1[12 * F_DIM];          // conv1 weights, K padded 9->12
    __shared__ float s_w2[F_DIM * F_DIM];       // conv2 weights
    __shared__ float s_wfc[2 * F_DIM * 16];     // fc1 slice for current frame (TDM dest)
    __shared__ float s_z[8 * 16 * ZLD];         // per-wave activation tiles
    __shared__ float s_mu[8 * 16];
    __shared__ float s_rs[8 * 16];
    __shared__ float s_lnw[F_DIM];
    __shared__ float s_lnb[F_DIM];

    const int tid  = threadIdx.x;
    const int wv   = tid >> 5;
    const int lane = tid & 31;
    const int nl   = lane & 15;     // column / row-in-halfwave index
    const int hi   = lane >> 4;     // half-wave select
    const int bidx = blockIdx.x;
    const int pixbase = blockIdx.y * 128 + wv * 16;
    const int zofs = wv * 16 * ZLD;

    // one-time weight staging
    for (int i = tid; i < 12 * F_DIM; i += 256) {
        int k = i / F_DIM, f = i - k * F_DIM;
        s_w1[i] = (k < C_DIM) ? w1[k * F_DIM + f] : 0.f;
    }
    for (int i = tid; i < F_DIM * F_DIM; i += 256) s_w2[i] = w2[i];
    if (tid < F_DIM) { s_lnw[tid] = lnw_c[tid]; s_lnb[tid] = lnb_c[tid]; }

    v8f accf = {};

    for (int tt = 0; tt < T_DIM; ++tt) {
        __syncthreads();   // everyone done with previous frame's s_wfc

#if __has_builtin(__builtin_amdgcn_tensor_load_to_lds) && \
    __has_builtin(__builtin_amdgcn_s_wait_tensorcnt)
        // ---- Tensor Data Mover: DMA the [128 rows x 16 floats] fc1 slice for
        // frame tt into LDS. Row fidx lives at wfc + (fidx*52 + tt)*16 floats,
        // i.e. a 2D tile: tile_dim0=16 elem, tile_dim1=128 rows, stride0=832 elem.
        if (wv == 0) {
            unsigned long long gaddr =
                (unsigned long long)(const void*)wfc + (unsigned long long)tt * 64ull;
            unsigned int ldsoff = (unsigned int)(unsigned long long)(void*)s_wfc;

            v4u g0;
            g0.x = 1u;                               // count=1, user descriptor
            g0.y = ldsoff;                           // lds_addr [63:32]
            g0.z = (unsigned int)(gaddr & 0xffffffffull);        // global_addr lo
            g0.w = (unsigned int)((gaddr >> 32) & 0x01ffffffull) // global_addr hi
                 | (2u << 30);                       // type = 2 ("image")

            v8i g1;
            g1[0] = (int)(2u << 16);                 // wg_mask=0, data_size=2 (4B)
            g1[1] = (int)(16u << 16);                // tensor_dim0=16 (lo16 @ bits 63:48)
            g1[2] = (int)(6656u << 16);              // dim0 hi=0 | tensor_dim1 lo16
            g1[3] = (int)(16u << 16);                // dim1 hi=0 | tile_dim0=16
            g1[4] = 128;                             // tile_dim1=128, tile_dim2=0
            g1[5] = 832;                             // tensor_dim0_stride lo32 = 832
            g1[6] = 0;                               // stride0 hi | stride1 lo
            g1[7] = 0;

            v4i gz = {0, 0, 0, 0};                   // groups 2/3 unused (2D tile)
#if __clang_major__ >= 23
            v8i gz8 = {0, 0, 0, 0, 0, 0, 0, 0};
            __builtin_amdgcn_tensor_load_to_lds(g0, g1, gz, gz, gz8, 0);
#else
            __builtin_amdgcn_tensor_load_to_lds(g0, g1, gz, gz, 0);
#endif
            __builtin_amdgcn_s_wait_tensorcnt(0);
        }
#else
        // fallback: cooperative vector load
        for (int i = tid; i < 2 * F_DIM * 16; i += 256) {
            int s = i >> 10, rem = i & 1023;
            int f = rem >> 4, j = rem & 15;
            s_wfc[i] = wfc[(((s * F_DIM + f) * T_DIM) + tt) * 16 + j];
        }
#endif
        __syncthreads();

        for (int src = 0; src < 2; ++src) {
            // ---- stage 1: input LayerNorm over channels (lanes 0..15, 1 pixel each)
            if (hi == 0) {
                int m  = nl;
                int pg = pixbase + m;
                int n  = bidx * HW + pg;
                float v[C_DIM];
                if (src == 0) {
                    #pragma unroll
                    for (int ch = 0; ch < C_DIM; ++ch)
                        v[ch] = x[((tt * B_DIM + bidx) * C_DIM + ch) * HW + pg];
                } else {
                    #pragma unroll
                    for (int ch = 0; ch < C_DIM; ++ch)
                        v[ch] = xs[(n * C_DIM + ch) * T_DIM + tt];
                }
                float mu = 0.f;
                #pragma unroll
                for (int ch = 0; ch < C_DIM; ++ch) mu += v[ch];
                mu *= (1.f / C_DIM);
                float var = 0.f;
                #pragma unroll
                for (int ch = 0; ch < C_DIM; ++ch) { float d = v[ch] - mu; var += d * d; }
                var *= (1.f / C_DIM);
                float rs = rsqrtf(var + LN_EPS);
                #pragma unroll
                for (int ch = 0; ch < C_DIM; ++ch)
                    s_z[zofs + m * ZLD + ch] = (v[ch] - mu) * rs * lnw_in[ch] + lnb_in[ch];
                s_z[zofs + m * ZLD + 9]  = 0.f;
                s_z[zofs + m * ZLD + 10] = 0.f;
                s_z[zofs + m * ZLD + 11] = 0.f;
            }
            // wave-private LDS; DS ops are in-order within a wave -> no barrier needed

            // ---- stage 2: conv1 9->64 via 3 K-steps of wmma f32 16x16x4
            v8f zf[4];
            #pragma unroll
            for (int ct = 0; ct < 4; ++ct) {
                v8f c = {};
                #pragma unroll
                for (int ks = 0; ks < 3; ++ks) {
                    int k0 = ks * 4 + hi * 2;
                    v2f a, bfr;
                    a.x   = s_z[zofs + nl * ZLD + k0];
                    a.y   = s_z[zofs + nl * ZLD + k0 + 1];
                    bfr.x = s_w1[(k0)     * F_DIM + ct * 16 + nl];
                    bfr.y = s_w1[(k0 + 1) * F_DIM + ct * 16 + nl];
                    c = wmma4(a, bfr, c);
                }
                zf[ct] = c;
            }
            #pragma unroll
            for (int ct = 0; ct < 4; ++ct) {
                #pragma unroll
                for (int r = 0; r < 8; ++r) {
                    int row = r + hi * 8, col = ct * 16 + nl;
                    s_z[zofs + row * ZLD + col] = fmaxf(zf[ct][r] + b1[col], 0.f);
                }
            }

            // ---- stage 3: conv2 64->64 via 16 K-steps per column tile
            #pragma unroll
            for (int ct = 0; ct < 4; ++ct) {
                v8f c = {};
                #pragma unroll
                for (int ks = 0; ks < 16; ++ks) {
                    int k0 = ks * 4 + hi * 2;
                    v2f a, bfr;
                    a.x   = s_z[zofs + nl * ZLD + k0];
                    a.y   = s_z[zofs + nl * ZLD + k0 + 1];
                    bfr.x = s_w2[(k0)     * F_DIM + ct * 16 + nl];
                    bfr.y = s_w2[(k0 + 1) * F_DIM + ct * 16 + nl];
                    c = wmma4(a, bfr, c);
                }
                zf[ct] = c;
            }
            #pragma unroll
            for (int ct = 0; ct < 4; ++ct) {
                #pragma unroll
                for (int r = 0; r < 8; ++r) {
                    int row = r + hi * 8, col = ct * 16 + nl;
                    s_z[zofs + row * ZLD + col] = fmaxf(zf[ct][r] + b2[col], 0.f);
                }
            }

            // ---- stage 4: LayerNorm stats over 64 features (lanes 0..15)
            if (hi == 0) {
                int m = nl;
                float s = 0.f;
                for (int f = 0; f < F_DIM; ++f) s += s_z[zofs + m * ZLD + f];
                float mu = s * (1.f / F_DIM);
                float var = 0.f;
                for (int f = 0; f < F_DIM; ++f) {
                    float d = s_z[zofs + m * ZLD + f] - mu;
                    var += d * d;
                }
                var *= (1.f / F_DIM);
                s_mu[wv * 16 + m] = mu;
                s_rs[wv * 16 + m] = rsqrtf(var + LN_EPS);
            }

            // ---- stage 5: fc1 slice accumulation, normalized-on-the-fly A operand
            float mu_r = s_mu[wv * 16 + nl];
            float rs_r = s_rs[wv * 16 + nl];
            #pragma unroll
            for (int ks = 0; ks < 16; ++ks) {
                int k0 = ks * 4 + hi * 2;
                v2f a, bfr;
                a.x = (s_z[zofs + nl * ZLD + k0]     - mu_r) * rs_r * s_lnw[k0]     + s_lnb[k0];
                a.y = (s_z[zofs + nl * ZLD + k0 + 1] - mu_r) * rs_r * s_lnw[k0 + 1] + s_lnb[k0 + 1];
                bfr.x = s_wfc[src * 1024 + (k0)     * 16 + nl];
                bfr.y = s_wfc[src * 1024 + (k0 + 1) * 16 + nl];
                accf = wmma4(a, bfr, accf);
            }
        }
    }

    // write per-pixel fc1 accumulator [N,16]
    #pragma unroll
    for (int r = 0; r < 8; ++r) {
        int row = r + hi * 8;
        int n   = bidx * HW + pixbase + row;
        acc[n * 16 + nl] = accf[r];
    }
}

// ---------------- kernel 5: fc head: relu(acc+b) @ fc2 + b -> tanh --------------
__global__ void k_head(const float* __restrict__ acc, const float* __restrict__ b1,
                       const float* __restrict__ w2, const float* __restrict__ b2,
                       float* __restrict__ theta_pred) {
    int n = blockIdx.x * blockDim.x + threadIdx.x;
    if (n >= N_PIX) return;
    float hbuf[16];
    #pragma unroll
    for (int j = 0; j < 16; ++j) hbuf[j] = fmaxf(acc[n * 16 + j] + b1[j], 0.f);
    #pragma unroll
    for (int k = 0; k < TH_D; ++k) {
        float s = b2[k];
        #pragma unroll
        for (int j = 0; j < 16; ++j) s += hbuf[j] * w2[j * TH_D + k];
        theta_pred[n * TH_D + k] = tanhf(s);
    }
}

// ---------------- launcher ------------------------------------------------------
extern "C" void kernel_launch(void* const* d_in, const int* in_sizes, int n_in,
                              void* d_out, int out_size, void* d_ws, size_t ws_size,
                              hipStream_t stream) {
    (void)in_sizes; (void)n_in; (void)out_size; (void)ws_size;

    const float* x      = (const float*)d_in[0];
    const float* traw   = (const float*)d_in[1];
    const float* lnw_in = (const float*)d_in[2];
    const float* lnb_in = (const float*)d_in[3];
    const float* w1     = (const float*)d_in[4];
    const float* b1     = (const float*)d_in[5];
    const float* w2     = (const float*)d_in[6];
    const float* b2     = (const float*)d_in[7];
    const float* lnw_c  = (const float*)d_in[8];
    const float* lnb_c  = (const float*)d_in[9];
    const float* wfc    = (const float*)d_in[10];
    const float* bfc    = (const float*)d_in[11];
    const float* w2h    = (const float*)d_in[12];
    const float* b2h    = (const float*)d_in[13];

    float* out       = (float*)d_out;
    float* o_aligned = out;                                   // [N,c,t]
    float* o_shift   = out + (size_t)N_PIX * C_DIM * T_DIM;   // [N,c,t]
    float* o_tpred   = o_shift + (size_t)N_PIX * C_DIM * T_DIM;
    float* o_tshift  = o_tpred + (size_t)N_PIX * TH_D;

    float* ws     = (float*)d_ws;
    float* bbuf   = ws;                        // 60 (pad 64)
    float* Ashift = ws + 64;                   // N*12
    float* Apred  = Ashift + N_PIX * 12;       // N*12
    float* accb   = Apred + N_PIX * 12;        // N*16

    k_basis<<<1, 32, 0, stream>>>(bbuf);
    k_theta2A<<<N_PIX / 256, 256, 0, stream>>>(traw, 0.5f, bbuf, Ashift, o_tshift);
    k_cpab<<<(N_PIX * T_DIM + 255) / 256, 256, 0, stream>>>(x, Ashift, o_shift);
    k_loc_fc<<<dim3(B_DIM, 32), 256, 0, stream>>>(x, o_shift, lnw_in, lnb_in,
                                                  w1, b1, w2, b2, lnw_c, lnb_c,
                                                  wfc, accb);
    k_head<<<N_PIX / 256, 256, 0, stream>>>(accb, bfc, w2h, b2h, o_tpred);
    k_theta2A<<<N_PIX / 256, 256, 0, stream>>>(o_tpred, 1.0f, bbuf, Apred, nullptr);
    k_cpab<<<(N_PIX * T_DIM + 255) / 256, 256, 0, stream>>>(x, Apred, o_aligned);
}